// Extra_51453708206798
// MI455X (gfx1250) — compile-verified
//
#include <hip/hip_runtime.h>
#include <hip/hip_bf16.h>
#include <math.h>

#define NTOK 200704
#define DIM 512
#define NCLS 4
#define NPROTO 8
#define NP 32
#define GAMMA_EMA 0.999f
#define INV_EPS_SINK 20.0f
#define TPB 128
#define NBLK_MAIN (NTOK / TPB)

typedef __attribute__((ext_vector_type(2))) float v2f;
typedef __attribute__((ext_vector_type(8))) float v8f;

// ---- workspace layout (float offsets) ----
#define WS_PN      0                      // 32*512 normalized prototypes
#define WS_PGT     (WS_PN + NP*DIM)       // [512][32] g .* Pn
#define WS_G1      (WS_PGT + DIM*NP)      // 32
#define WS_B1      (WS_G1 + NP)           // 32
#define WS_CONST   (WS_B1 + NP)           // 4: sum g^2, sum g*b, sum b^2
#define WS_S0      (WS_CONST + 4)         // 4 per-class total exp mass
#define WS_BCNT    (WS_S0 + NCLS)         // 4 per-class token counts
#define WS_SA      (WS_BCNT + NCLS)       // 32 sinkhorn col-sum buffer A
#define WS_SB      (WS_SA + NP)           // 32 sinkhorn col-sum buffer B
#define WS_C       (WS_SB + NP)           // 32 column scale factors
#define WS_NCNT    (WS_C + NP)            // 32 per-prototype counts
#define WS_FACC    (WS_NCNT + NP)         // 32*512 f accumulator
#define WS_E8      (WS_FACC + NP*DIM)     // NTOK*8 exp scores (own class)
#define WS_R       (WS_E8 + (size_t)NTOK*8) // NTOK row scale factors
#define WS_CORR    (WS_R + NTOK)          // NTOK bytes (pred==gt flags)

// ================= K0: setup =================
__global__ void __launch_bounds__(256)
k0_setup(const float* __restrict__ protos, const float* __restrict__ g,
         const float* __restrict__ b, float* __restrict__ ws) {
  __shared__ float red[256];
  const int bid = blockIdx.x, t = threadIdx.x;
  if (bid < NP) {
    const float* pr = protos + bid * DIM;
    float s = 0.f;
    for (int d = t; d < DIM; d += 256) { float v = pr[d]; s += v * v; }
    red[t] = s; __syncthreads();
    for (int o = 128; o > 0; o >>= 1) { if (t < o) red[t] += red[t + o]; __syncthreads(); }
    const float inv = 1.f / fmaxf(sqrtf(red[0]), 1e-12f);
    __syncthreads();
    float sg = 0.f, sb = 0.f;
    for (int d = t; d < DIM; d += 256) {
      const float pn = pr[d] * inv;
      ws[WS_PN + bid * DIM + d] = pn;
      ws[WS_PGT + d * NP + bid] = g[d] * pn;
      sg += g[d] * pn; sb += b[d] * pn;
    }
    red[t] = sg; __syncthreads();
    for (int o = 128; o > 0; o >>= 1) { if (t < o) red[t] += red[t + o]; __syncthreads(); }
    if (t == 0) ws[WS_G1 + bid] = red[0];
    __syncthreads();
    red[t] = sb; __syncthreads();
    for (int o = 128; o > 0; o >>= 1) { if (t < o) red[t] += red[t + o]; __syncthreads(); }
    if (t == 0) ws[WS_B1 + bid] = red[0];
  } else if (bid == NP) {
    float s2 = 0.f, sgb = 0.f, sbb = 0.f;
    for (int d = t; d < DIM; d += 256) { float gg = g[d], bb = b[d]; s2 += gg*gg; sgb += gg*bb; sbb += bb*bb; }
    red[t] = s2; __syncthreads();
    for (int o = 128; o > 0; o >>= 1) { if (t < o) red[t] += red[t + o]; __syncthreads(); }
    if (t == 0) ws[WS_CONST + 0] = red[0];
    __syncthreads();
    red[t] = sgb; __syncthreads();
    for (int o = 128; o > 0; o >>= 1) { if (t < o) red[t] += red[t + o]; __syncthreads(); }
    if (t == 0) ws[WS_CONST + 1] = red[0];
    __syncthreads();
    red[t] = sbb; __syncthreads();
    for (int o = 128; o > 0; o >>= 1) { if (t < o) red[t] += red[t + o]; __syncthreads(); }
    if (t == 0) ws[WS_CONST + 2] = red[0];
    if (t < NCLS) { ws[WS_S0 + t] = 0.f; ws[WS_BCNT + t] = 0.f; }
    if (t < NP)   { ws[WS_SA + t] = 0.f; ws[WS_SB + t] = 0.f; ws[WS_C + t] = 1.f; ws[WS_NCNT + t] = 0.f; }
  } else {
    const int i = (bid - NP - 1) * 256 + t;
    if (i < NP * DIM) ws[WS_FACC + i] = 0.f;
  }
}

// ================= K1: main streaming pass (WMMA) =================
__global__ void __launch_bounds__(256)
k1_main(const float* __restrict__ feats, const float* __restrict__ g,
        const float* __restrict__ b, const float* __restrict__ mg,
        const float* __restrict__ mb, const int* __restrict__ gt,
        float* __restrict__ ws, float* __restrict__ out,
        unsigned char* __restrict__ corr) {
  __shared__ float PG[DIM * 40];       // B matrix (g .* Pn)^T, padded stride 40
  __shared__ float XS[TPB * 68];       // token slab, padded stride 68
  __shared__ float SC[TPB * 36];       // raw GX scores per token
  __shared__ float AUX[TPB * 3];
  __shared__ float G2S[64], GBS[64];
  __shared__ float G1S[NP], B1S[NP];
  __shared__ float lS1[NP], lS0[NCLS], lB[NCLS];

  const int t = threadIdx.x;
  const int wv = t >> 5, lane = t & 31;
  const int n0 = blockIdx.x * TPB;
  const int mrow = lane & 15;          // A-matrix M (token in tile) / B-matrix N
  const int khalf = lane >> 4;
  const int tloc = wv * 16 + mrow;

  for (int i = t; i < DIM * NP; i += 256)
    PG[(i >> 5) * 40 + (i & 31)] = ws[WS_PGT + i];
  if (t < NP) { lS1[t] = 0.f; G1S[t] = ws[WS_G1 + t]; B1S[t] = ws[WS_B1 + t]; }
  if (t < NCLS) { lS0[t] = 0.f; lB[t] = 0.f; }

  v8f acc0 = {}, acc1 = {};
  float sx = 0.f, sxx = 0.f, sg2x = 0.f, sg2xx = 0.f, sgbx = 0.f;

  for (int kc = 0; kc < DIM; kc += 64) {
    __syncthreads();
    for (int i = 0; i < 8; ++i) {
      const int v = t + 256 * i;
      const int row = v >> 4, c4 = (v & 15) << 2;
      const float4 s4 = *(const float4*)(feats + (size_t)(n0 + row) * DIM + kc + c4);
      float* dst = &XS[row * 68 + c4];
      dst[0] = s4.x; dst[1] = s4.y; dst[2] = s4.z; dst[3] = s4.w;
    }
    if (t < 64) { const float gg = g[kc + t]; G2S[t] = gg * gg; GBS[t] = gg * b[kc + t]; }
    __syncthreads();
    {   // aux accumulation: 2 lanes per token split even/odd columns
      const float* xr = &XS[tloc * 68];
      for (int c = khalf; c < 64; c += 2) {
        const float x = xr[c], g2 = G2S[c], gb = GBS[c];
        sx += x; sxx += x * x; sg2x += g2 * x; sg2xx += g2 * x * x; sgbx += gb * x;
      }
    }
    // WMMA f32 16x16x4: D(tokens x protos) += A(tokens x 4) * B(4 x protos)
    for (int k4 = 0; k4 < 64; k4 += 4) {
      const int kk = k4 + 2 * khalf;
      v2f a; a.x = XS[tloc * 68 + kk]; a.y = XS[tloc * 68 + kk + 1];
      const float* pg0 = &PG[kk * 40];
      const float* pg1 = &PG[(kk + 1) * 40];
      v2f b0; b0.x = pg0[mrow];      b0.y = pg1[mrow];
      v2f b1; b1.x = pg0[16 + mrow]; b1.y = pg1[16 + mrow];
      acc0 = __builtin_amdgcn_wmma_f32_16x16x4_f32(false, a, false, b0, (short)0, acc0, false, false);
      acc1 = __builtin_amdgcn_wmma_f32_16x16x4_f32(false, a, false, b1, (short)0, acc1, false, false);
    }
  }
  // combine lane-pair aux partials
  sx    += __shfl_xor(sx, 16, 32);
  sxx   += __shfl_xor(sxx, 16, 32);
  sg2x  += __shfl_xor(sg2x, 16, 32);
  sg2xx += __shfl_xor(sg2xx, 16, 32);
  sgbx  += __shfl_xor(sgbx, 16, 32);
  if (khalf == 0) {
    const float mu = sx * (1.f / DIM);
    const float var = sxx * (1.f / DIM) - mu * mu;
    const float is = rsqrtf(var + 1e-5f);
    const float sg2c = ws[WS_CONST + 0], sgbc = ws[WS_CONST + 1], sbbc = ws[WS_CONST + 2];
    const float ny2 = is * is * (sg2xx - 2.f * mu * sg2x + mu * mu * sg2c)
                    + 2.f * is * (sgbx - mu * sgbc) + sbbc;
    const float iy = 1.f / fmaxf(sqrtf(fmaxf(ny2, 0.f)), 1e-12f);
    AUX[tloc * 3 + 0] = mu; AUX[tloc * 3 + 1] = is; AUX[tloc * 3 + 2] = iy;
  }
  // dump accumulator tiles: VGPR r -> token M=r (lanes 0-15) / r+8 (lanes 16-31)
  const int trow = wv * 16 + 8 * khalf;
  #pragma unroll
  for (int r = 0; r < 8; ++r) {
    SC[(trow + r) * 36 + mrow]      = acc0[r];
    SC[(trow + r) * 36 + 16 + mrow] = acc1[r];
  }
  __syncthreads();
  if (t < TPB) {
    const int n = n0 + t;
    const float mu = AUX[t * 3], is = AUX[t * 3 + 1], iy = AUX[t * 3 + 2];
    float s[NP];
    #pragma unroll
    for (int p = 0; p < NP; ++p)
      s[p] = iy * (is * (SC[t * 36 + p] - mu * G1S[p]) + B1S[p]);
    float mx[NCLS]; float mean4 = 0.f;
    #pragma unroll
    for (int k = 0; k < NCLS; ++k) {
      float m0 = s[k * NPROTO];
      #pragma unroll
      for (int m = 1; m < NPROTO; ++m) m0 = fmaxf(m0, s[k * NPROTO + m]);
      mx[k] = m0; mean4 += m0;
    }
    mean4 *= 0.25f;
    float var4 = 0.f;
    #pragma unroll
    for (int k = 0; k < NCLS; ++k) { const float d = mx[k] - mean4; var4 += d * d; }
    var4 *= 0.25f;
    const float inv4 = rsqrtf(var4 + 1e-5f);
    int pred = 0; float best = -3.402823466e38f;
    #pragma unroll
    for (int k = 0; k < NCLS; ++k) {
      const float o = (mx[k] - mean4) * inv4 * mg[k] + mb[k];
      out[(size_t)n * NCLS + k] = o;
      if (o > best) { best = o; pred = k; }
    }
    const int kcls = gt[n];
    float esum = 0.f;
    #pragma unroll
    for (int m = 0; m < NPROTO; ++m) {
      const float e = expf(INV_EPS_SINK * s[kcls * NPROTO + m]);
      ws[WS_E8 + (size_t)n * NPROTO + m] = e;
      esum += e;
      atomicAdd(&lS1[kcls * NPROTO + m], e);
    }
    atomicAdd(&lS0[kcls], esum);
    atomicAdd(&lB[kcls], 1.f);
    corr[n] = (unsigned char)(pred == kcls);
  }
  __syncthreads();
  if (t < NP) atomicAdd(&ws[WS_SA + t], lS1[t]);
  if (t < NCLS) { atomicAdd(&ws[WS_S0 + t], lS0[t]); atomicAdd(&ws[WS_BCNT + t], lB[t]); }
}

// ============ K2a: column normalization (tiny) ============
__global__ void k2a_col(float* __restrict__ ws, int sInOff, int sZeroOff, int first) {
  const int m = threadIdx.x;
  if (m < NP) {
    const float s = ws[sInOff + m];
    const float cp = ws[WS_C + m];
    const float scale = first ? (1.f / fmaxf(ws[WS_S0 + m / NPROTO], 1e-30f)) : 1.f;
    const float col = cp * s * scale;
    ws[WS_C + m] = cp / (fmaxf(col, 1e-30f) * (float)NPROTO);
    if (sZeroOff >= 0) ws[sZeroOff + m] = 0.f;
  }
}

// ============ K2b: row normalization + next column sums ============
__global__ void __launch_bounds__(256)
k2b_row(const int* __restrict__ gt, float* __restrict__ ws, int sOutOff, int first) {
  __shared__ float cL[NP], lS[NP];
  const int t = threadIdx.x;
  if (t < NP) { cL[t] = ws[WS_C + t]; lS[t] = 0.f; }
  __syncthreads();
  const int n = blockIdx.x * 256 + t;
  const int k = gt[n];
  const float* ep = &ws[WS_E8 + (size_t)n * NPROTO];
  float e[NPROTO]; float T = 0.f;
  #pragma unroll
  for (int m = 0; m < NPROTO; ++m) { e[m] = ep[m]; T += e[m] * cL[k * NPROTO + m]; }
  const float Bk = fmaxf(ws[WS_BCNT + k], 1.f);
  const float rp = first ? (1.f / fmaxf(ws[WS_S0 + k], 1e-30f)) : ws[WS_R + n];
  const float row = rp * T;
  const float r = rp / (fmaxf(row, 1e-30f) * Bk);
  ws[WS_R + n] = r;
  #pragma unroll
  for (int m = 0; m < NPROTO; ++m) atomicAdd(&lS[k * NPROTO + m], e[m] * r);
  __syncthreads();
  if (t < NP) atomicAdd(&ws[sOutOff + t], lS[t]);
}

// ============ K3: second streaming pass — prototype aggregation ============
__global__ void __launch_bounds__(256)
k3_agg(const float* __restrict__ feats, const float* __restrict__ g,
       const float* __restrict__ b, const int* __restrict__ gt,
       const unsigned char* __restrict__ corr, float* __restrict__ ws) {
  __shared__ float XS[TPB * 68];
  __shared__ float FA[NP * 516];       // padded stride 516 -> bank spread
  __shared__ float GS[64], BS[64];
  __shared__ float AUXm[TPB], AUXs[TPB], AUXy[TPB];
  __shared__ int ROWL[TPB];
  __shared__ float lN[NP];

  const int t = threadIdx.x;
  const int n0 = blockIdx.x * TPB;
  for (int i = t; i < NP * 516; i += 256) FA[i] = 0.f;
  if (t < NP) lN[t] = 0.f;

  float sx = 0.f, sxx = 0.f, sg2x = 0.f, sg2xx = 0.f, sgbx = 0.f;
  for (int kc = 0; kc < DIM; kc += 64) {
    __syncthreads();
    for (int i = 0; i < 8; ++i) {
      const int v = t + 256 * i;
      const int row = v >> 4, c4 = (v & 15) << 2;
      const float4 s4 = *(const float4*)(feats + (size_t)(n0 + row) * DIM + kc + c4);
      float* dst = &XS[row * 68 + c4];
      dst[0] = s4.x; dst[1] = s4.y; dst[2] = s4.z; dst[3] = s4.w;
    }
    if (t < 64) { GS[t] = g[kc + t]; BS[t] = b[kc + t]; }
    __syncthreads();
    if (t < TPB) {
      const float* xr = &XS[t * 68];
      for (int c = 0; c < 64; ++c) {
        const float x = xr[c], gg = GS[c], bb = BS[c];
        sx += x; sxx += x * x; sg2x += gg * gg * x; sg2xx += gg * gg * x * x; sgbx += gg * bb * x;
      }
    }
  }
  __syncthreads();
  if (t < TPB) {
    const float mu = sx * (1.f / DIM);
    const float var = sxx * (1.f / DIM) - mu * mu;
    const float is = rsqrtf(var + 1e-5f);
    const float sg2c = ws[WS_CONST + 0], sgbc = ws[WS_CONST + 1], sbbc = ws[WS_CONST + 2];
    const float ny2 = is * is * (sg2xx - 2.f * mu * sg2x + mu * mu * sg2c)
                    + 2.f * is * (sgbx - mu * sgbc) + sbbc;
    const float iy = 1.f / fmaxf(sqrtf(fmaxf(ny2, 0.f)), 1e-12f);
    AUXm[t] = mu; AUXs[t] = is; AUXy[t] = iy;
    const int n = n0 + t;
    const int k = gt[n];
    int rowi = -1;
    if (corr[n]) {
      const float* ep = &ws[WS_E8 + (size_t)n * NPROTO];
      float bestv = -1.f; int besti = 0;
      #pragma unroll
      for (int m = 0; m < NPROTO; ++m) {
        const float v = ep[m] * ws[WS_C + k * NPROTO + m];
        if (v > bestv) { bestv = v; besti = m; }
      }
      rowi = k * NPROTO + besti;
      atomicAdd(&lN[rowi], 1.f);
    }
    ROWL[t] = rowi;
  }
  // second pass over columns: accumulate normalized rows into FA
  const int tt = t >> 1, half = t & 1;
  for (int kc = 0; kc < DIM; kc += 64) {
    __syncthreads();
    for (int i = 0; i < 8; ++i) {
      const int v = t + 256 * i;
      const int row = v >> 4, c4 = (v & 15) << 2;
      const float4 s4 = *(const float4*)(feats + (size_t)(n0 + row) * DIM + kc + c4);
      float* dst = &XS[row * 68 + c4];
      dst[0] = s4.x; dst[1] = s4.y; dst[2] = s4.z; dst[3] = s4.w;
    }
    if (t < 64) { GS[t] = g[kc + t]; BS[t] = b[kc + t]; }
    __syncthreads();
    const int rowi = ROWL[tt];
    if (rowi >= 0) {
      const float mu = AUXm[tt], is = AUXs[tt], iy = AUXy[tt];
      const float* xr = &XS[tt * 68];
      float* fr = &FA[rowi * 516 + kc];
      for (int c = half * 32; c < half * 32 + 32; ++c)
        atomicAdd(&fr[c], iy * (is * GS[c] * (xr[c] - mu) + BS[c]));
    }
  }
  __syncthreads();
  for (int i = t; i < NP * DIM; i += 256) {
    const float v = FA[(i >> 9) * 516 + (i & 511)];
    if (v != 0.f) atomicAdd(&ws[WS_FACC + i], v);
  }
  if (t < NP) { const float v = lN[t]; if (v != 0.f) atomicAdd(&ws[WS_NCNT + t], v); }
}

// ============ K4: EMA update + renormalize ============
__global__ void __launch_bounds__(256)
k4_fin(float* __restrict__ ws, float* __restrict__ outp) {
  __shared__ float red[256];
  __shared__ float frow[DIM];
  const int p = blockIdx.x, t = threadIdx.x;
  const int k = p / NPROTO;
  float s = 0.f;
  for (int d = t; d < DIM; d += 256) { const float v = ws[WS_FACC + p * DIM + d]; frow[d] = v; s += v * v; }
  red[t] = s; __syncthreads();
  for (int o = 128; o > 0; o >>= 1) { if (t < o) red[t] += red[t + o]; __syncthreads(); }
  const float invf = 1.f / fmaxf(sqrtf(red[0]), 1e-12f);
  const float ncp = ws[WS_NCNT + p];
  float nsum = 0.f;
  for (int m = 0; m < NPROTO; ++m) nsum += ws[WS_NCNT + k * NPROTO + m];
  const float selc = ws[WS_BCNT + k];
  const bool valid = (ncp != 0.f) && (selc > 0.f) && (nsum > 0.f);
  __syncthreads();
  float s2 = 0.f;
  for (int d = t; d < DIM; d += 256) {
    const float pn = ws[WS_PN + p * DIM + d];
    const float upd = GAMMA_EMA * pn + (1.f - GAMMA_EMA) * (frow[d] * invf);
    const float nv = valid ? upd : pn;
    frow[d] = nv; s2 += nv * nv;
  }
  __syncthreads();
  red[t] = s2; __syncthreads();
  for (int o = 128; o > 0; o >>= 1) { if (t < o) red[t] += red[t + o]; __syncthreads(); }
  const float invn = 1.f / fmaxf(sqrtf(red[0]), 1e-12f);
  for (int d = t; d < DIM; d += 256) outp[p * DIM + d] = frow[d] * invn;
}

extern "C" void kernel_launch(void* const* d_in, const int* in_sizes, int n_in,
                              void* d_out, int out_size, void* d_ws, size_t ws_size,
                              hipStream_t stream) {
  const float* feats  = (const float*)d_in[0];
  const float* ln_g   = (const float*)d_in[1];
  const float* ln_b   = (const float*)d_in[2];
  const float* mg     = (const float*)d_in[3];
  const float* mb     = (const float*)d_in[4];
  const float* protos = (const float*)d_in[5];
  const int*   gt     = (const int*)d_in[6];
  float* out = (float*)d_out;
  float* ws  = (float*)d_ws;
  unsigned char* corr = (unsigned char*)(ws + WS_CORR);

  k0_setup<<<NP + 1 + 64, 256, 0, stream>>>(protos, ln_g, ln_b, ws);
  k1_main<<<NBLK_MAIN, 256, 0, stream>>>(feats, ln_g, ln_b, mg, mb, gt, ws, out, corr);
  // Sinkhorn: 3 iterations, factorized as col-scale (k2a) + row-scale/colsum pass (k2b)
  k2a_col<<<1, 32, 0, stream>>>(ws, WS_SA, WS_SB, 1);
  k2b_row<<<NTOK / 256, 256, 0, stream>>>(gt, ws, WS_SB, 1);
  k2a_col<<<1, 32, 0, stream>>>(ws, WS_SB, WS_SA, 0);
  k2b_row<<<NTOK / 256, 256, 0, stream>>>(gt, ws, WS_SA, 0);
  k2a_col<<<1, 32, 0, stream>>>(ws, WS_SA, -1, 0);
  k3_agg<<<NBLK_MAIN, 256, 0, stream>>>(feats, ln_g, ln_b, gt, corr, ws);
  k4_fin<<<NP, 256, 0, stream>>>(ws, out + (size_t)NTOK * NCLS);
}